// CustomNetworkGINSeroMean_90778428768945
// MI455X (gfx1250) — compile-verified
//
#include <hip/hip_runtime.h>
#include <math.h>

// ---------------------------------------------------------------------------
// CDNA5 / gfx1250 implementation of the GIN+SERO network (fp32 end-to-end).
// Matrix work goes through V_WMMA_F32_16X16X4_F32 (wave32, 16x16 tiles, K=4).
// ---------------------------------------------------------------------------

typedef float v2f __attribute__((ext_vector_type(2)));
typedef float v8f __attribute__((ext_vector_type(8)));

__device__ __forceinline__ v8f wmma_f32x4(v2f a, v2f b, v8f c) {
  // (neg_a, A, neg_b, B, c_mod, C, reuse_a, reuse_b)
  return __builtin_amdgcn_wmma_f32_16x16x4_f32(false, a, false, b, (short)0, c,
                                               false, false);
}

// A-frag (16x4 f32): lane L: m = L&15; VGPR0 -> K = 2*(L>>4), VGPR1 -> K+1.
// B-frag (4x16 f32): lane L: n = L&15; VGPR0 -> K = 2*(L>>4), VGPR1 -> K+1.
// C/D (16x16 f32):   lane L, vgpr v: row = v + 8*(L>>4), col = L&15.

#define WAVE_SETUP()                                  \
  const int lane = threadIdx.x & 31;                  \
  const int wave = threadIdx.x >> 5;                  \
  const int half = lane >> 4;                         \
  const int lm = lane & 15;

// ---------------------------------------------------------------------------
// Kernel 1: batched Y = adj @ x   (per batch: (n x n) @ (n x IN)).
// Edge handling: clamped (always in-bounds) addresses + MULTIPLICATIVE 0/1
// masks. The multiply keeps every load live on all paths, so the compiler
// cannot sink loads under EXEC branches -> branch-free inner loop.
// ---------------------------------------------------------------------------
__global__ void k_adjx_wmma(const float* __restrict__ adj,
                            const float* __restrict__ x,
                            float* __restrict__ y, int n, int IN) {
  WAVE_SETUP();
  const int M = n, N = IN, K = n;
  const int tiles_n = (N + 15) >> 4;
  const int tiles_m = (M + 15) >> 4;
  const int tid = blockIdx.x * 8 + wave;
  if (tid >= tiles_m * tiles_n) return;
  const int tm = tid / tiles_n, tn = tid % tiles_n;
  const int b = blockIdx.y;
  const float* A = adj + (size_t)b * n * n;
  const float* Bm = x + (size_t)b * n * IN;
  float* Y = y + (size_t)b * n * IN;

  v8f acc;
  for (int v = 0; v < 8; ++v) acc[v] = 0.0f;
  const int ar = tm * 16 + lm;
  const int bc = tn * 16 + lm;
  const float am = (ar < M) ? 1.0f : 0.0f;   // row mask (A)
  const float bm = (bc < N) ? 1.0f : 0.0f;   // col mask (B)
  const int arc = (ar < M) ? ar : (M - 1);
  const int bcc = (bc < N) ? bc : (N - 1);
  const float* Arow = A + (size_t)arc * K;
  for (int kb = 0; kb < K; kb += 4) {
    const int k0 = kb + 2 * half;
    const int k1 = k0 + 1;
    const float km0 = (k0 < K) ? 1.0f : 0.0f;
    const float km1 = (k1 < K) ? 1.0f : 0.0f;
    const int k0c = (k0 < K) ? k0 : (K - 1);
    const int k1c = (k1 < K) ? k1 : (K - 1);
    v2f a, bf;
    a.x = Arow[k0c] * (am * km0);
    a.y = Arow[k1c] * (am * km1);
    bf.x = Bm[(size_t)k0c * N + bcc] * (bm * km0);
    bf.y = Bm[(size_t)k1c * N + bcc] * (bm * km1);
    acc = wmma_f32x4(a, bf, acc);
  }
  for (int v = 0; v < 8; ++v) {
    const int r = tm * 16 + v + 8 * half;
    const int c = tn * 16 + lm;
    if (r < M && c < N) Y[(size_t)r * N + c] = acc[v];
  }
}

// ---------------------------------------------------------------------------
// Kernel 2: h = x + (adj@x)/deg, deg = max(rowsum(adj), 1e-12). 1 block/node.
// ---------------------------------------------------------------------------
__global__ void k_degh(const float* __restrict__ adj,
                       const float* __restrict__ x,
                       const float* __restrict__ y, float* __restrict__ h,
                       int n, int IN) {
  const int r = blockIdx.x;          // global node index b*n + node
  const int b = r / n, node = r % n;
  const int tid = threadIdx.x;
  __shared__ float red[128];
  const float* arow = adj + ((size_t)b * n + node) * n;
  float s = 0.0f;
  for (int j = tid; j < n; j += 128) s += arow[j];
  red[tid] = s;
  __syncthreads();
  for (int off = 64; off > 0; off >>= 1) {
    if (tid < off) red[tid] += red[tid + off];
    __syncthreads();
  }
  const float deg = fmaxf(red[0], 1e-12f);
  for (int c = tid; c < IN; c += 128) {
    const size_t o = (size_t)r * IN + c;
    h[o] = x[o] + y[o] / deg;
  }
}

// ---------------------------------------------------------------------------
// Kernel 3: Apos9[r,k] = relu(pos @ w1)[r,k] for k<8 ; = 1.0 for k==8.
// The ones-column folds the bias block of the GIN GEMM into a uniform form.
// ---------------------------------------------------------------------------
__global__ void k_posw1(const float* __restrict__ pos,
                        const float* __restrict__ w1,
                        float* __restrict__ apos9, int total) {
  const int idx = blockIdx.x * 256 + threadIdx.x;
  if (idx >= total) return;
  const int r = idx / 9, k = idx % 9;
  if (k == 8) {
    apos9[idx] = 1.0f;
    return;
  }
  const float* prow = pos + (size_t)r * 100;
  float s = 0.0f;
  for (int j = 0; j < 100; ++j) s += prow[j] * w1[j * 8 + k];
  apos9[idx] = fmaxf(s, 0.0f);
}

// ---------------------------------------------------------------------------
// Kernel 3b: Waug = concat(w2 (8*IN x 64), b2 (IN x 64))  (tiny copy)
// ---------------------------------------------------------------------------
__global__ void k_waug(const float* __restrict__ w2, const float* __restrict__ b2,
                       float* __restrict__ waug, int KI64, int total) {
  const int idx = blockIdx.x * 256 + threadIdx.x;
  if (idx >= total) return;
  waug[idx] = (idx < KI64) ? w2[idx] : b2[idx - KI64];
}

// ---------------------------------------------------------------------------
// Kernel 4: GIN fused GEMM, fully branch-free.
//   xo[r,o] = sum_{k<9} sum_{i<IN} Apos9[r,k]*h[r,i] * Waug[k*IN+i, o]
// One wave computes a 16x64 strip (4 accumulators, A-fragment reused 4x).
// M = B*n is a multiple of 16; IN is a multiple of 4 -> no guards anywhere.
// ---------------------------------------------------------------------------
template <int IN>
__global__ void k_ginmm_wmma(const float* __restrict__ apos9,
                             const float* __restrict__ h,
                             const float* __restrict__ waug,
                             float* __restrict__ xo, int tiles_m) {
  WAVE_SETUP();
  const int tm = blockIdx.x * 8 + wave;
  if (tm >= tiles_m) return;

  v8f acc0, acc1, acc2, acc3;
  for (int v = 0; v < 8; ++v) {
    acc0[v] = 0.0f; acc1[v] = 0.0f; acc2[v] = 0.0f; acc3[v] = 0.0f;
  }
  const int ar = tm * 16 + lm;               // row owned by this lane's A-frag
  const float* hrow = h + (size_t)ar * IN;
  const float* ap = apos9 + (size_t)ar * 9;

  for (int k = 0; k < 9; ++k) {
    const float ak = ap[k];
    const float* wblk = waug + (size_t)k * IN * 64;
#pragma unroll 4
    for (int ib = 0; ib < IN; ib += 4) {
      const int i0 = ib + 2 * half;
      v2f a;
      a.x = ak * hrow[i0];
      a.y = ak * hrow[i0 + 1];
      const float* w0 = wblk + (size_t)i0 * 64;        // row i0
      const float* w1 = w0 + 64;                       // row i0+1
      v2f bf0, bf1, bf2, bf3;
      bf0.x = w0[lm];       bf0.y = w1[lm];
      bf1.x = w0[16 + lm];  bf1.y = w1[16 + lm];
      bf2.x = w0[32 + lm];  bf2.y = w1[32 + lm];
      bf3.x = w0[48 + lm];  bf3.y = w1[48 + lm];
      acc0 = wmma_f32x4(a, bf0, acc0);
      acc1 = wmma_f32x4(a, bf1, acc1);
      acc2 = wmma_f32x4(a, bf2, acc2);
      acc3 = wmma_f32x4(a, bf3, acc3);
    }
  }
  float* orow = xo + (size_t)(tm * 16 + 8 * half) * 64 + lm;
  for (int v = 0; v < 8; ++v) {
    orow[(size_t)v * 64 + 0]  = acc0[v];
    orow[(size_t)v * 64 + 16] = acc1[v];
    orow[(size_t)v * 64 + 32] = acc2[v];
    orow[(size_t)v * 64 + 48] = acc3[v];
  }
}

// ---------------------------------------------------------------------------
// Kernel 5: top-k (exact rank, stable tie-break == jax.lax.top_k order)
// ---------------------------------------------------------------------------
__global__ void k_topk(const float* __restrict__ xo,
                       const float* __restrict__ pw, int* __restrict__ perm,
                       float* __restrict__ vals, int n, int m) {
  const int b = blockIdx.x, tid = threadIdx.x;
  __shared__ float sc[128];
  __shared__ float invn;
  if (tid == 0) {
    float s = 0.0f;
    for (int o = 0; o < 64; ++o) s += pw[o] * pw[o];
    invn = 1.0f / sqrtf(s);
  }
  __syncthreads();
  if (tid < n) {
    const float* row = xo + ((size_t)b * n + tid) * 64;
    float d = 0.0f;
    for (int o = 0; o < 64; ++o) d += row[o] * pw[o];
    const float t = d * invn;
    sc[tid] = 1.0f / (1.0f + expf(-t));
  }
  __syncthreads();
  if (tid < n) {
    const float v = sc[tid];
    int rank = 0;
    for (int t = 0; t < n; ++t) {
      const float u = sc[t];
      rank += (u > v) || (u == v && t < tid);
    }
    if (rank < m) {
      perm[b * m + rank] = tid;
      vals[b * m + rank] = v;
    }
  }
}

// ---------------------------------------------------------------------------
// Kernel 6: gather pooled x (scaled by vals), adj (both axes), pos (rows)
// ---------------------------------------------------------------------------
__global__ void k_gather(const float* __restrict__ xo,
                         const float* __restrict__ adj,
                         const float* __restrict__ pos,
                         const int* __restrict__ perm,
                         const float* __restrict__ vals,
                         float* __restrict__ xp, float* __restrict__ adp,
                         float* __restrict__ pp, int n, int m) {
  const int g = blockIdx.x;
  const int b = g / m, r = g % m;
  const int tid = threadIdx.x;
  const int p = perm[b * m + r];
  const float v = vals[b * m + r];
  const float* xsrc = xo + ((size_t)b * n + p) * 64;
  float* xdst = xp + ((size_t)b * m + r) * 64;
  for (int o = tid; o < 64; o += 128) xdst[o] = xsrc[o] * v;
  const float* arow = adj + ((size_t)b * n + p) * n;
  float* adst = adp + ((size_t)b * m + r) * m;
  for (int c = tid; c < m; c += 128) adst[c] = arow[perm[b * m + c]];
  const float* prow = pos + ((size_t)b * n + p) * 100;
  float* pdst = pp + ((size_t)b * m + r) * 100;
  for (int c = tid; c < 100; c += 128) pdst[c] = prow[c];
}

// ---------------------------------------------------------------------------
// Kernel 7: r = mean over nodes of pooled x
// ---------------------------------------------------------------------------
__global__ void k_rmean(const float* __restrict__ xp, float* __restrict__ rb,
                        int m) {
  const int b = blockIdx.x, o = threadIdx.x;  // 64 threads
  float s = 0.0f;
  for (int r = 0; r < m; ++r) s += xp[((size_t)b * m + r) * 64 + o];
  rb[b * 64 + o] = s / (float)m;
}

// ---------------------------------------------------------------------------
// Kernel 8: SERO readout.  (x*att).mean(1) == r * att.
//   z = r@sew+seb ; z = BN(z) (batch stats) ; z = gelu(z) (exact erf)
//   att = sigmoid(z@saw+sab) ; zcat[:, 64*layer:...] = r*att
// ---------------------------------------------------------------------------
__global__ void k_sero(const float* __restrict__ rb,
                       const float* __restrict__ sew,
                       const float* __restrict__ seb,
                       const float* __restrict__ sbg,
                       const float* __restrict__ sbb,
                       const float* __restrict__ saw,
                       const float* __restrict__ sab, float* __restrict__ zc,
                       int layer) {
  const int tid = threadIdx.x;  // 256 threads, 1 block
  __shared__ float Z[64 * 64];
  __shared__ float mu[64], iv[64];
  for (int idx = tid; idx < 4096; idx += 256) {
    const int bq = idx >> 6, o = idx & 63;
    float s = seb[o];
    for (int j = 0; j < 64; ++j) s += rb[bq * 64 + j] * sew[j * 64 + o];
    Z[idx] = s;
  }
  __syncthreads();
  if (tid < 64) {
    float s = 0.0f, s2 = 0.0f;
    for (int bq = 0; bq < 64; ++bq) {
      const float z = Z[bq * 64 + tid];
      s += z;
      s2 += z * z;
    }
    const float mn = s * (1.0f / 64.0f);
    const float var = s2 * (1.0f / 64.0f) - mn * mn;
    mu[tid] = mn;
    iv[tid] = 1.0f / sqrtf(var + 1e-5f);
  }
  __syncthreads();
  for (int idx = tid; idx < 4096; idx += 256) {
    const int o = idx & 63;
    const float zz = (Z[idx] - mu[o]) * iv[o] * sbg[o] + sbb[o];
    Z[idx] = 0.5f * zz * (1.0f + erff(zz * 0.70710678118654752f));
  }
  __syncthreads();
  for (int idx = tid; idx < 4096; idx += 256) {
    const int bq = idx >> 6, o = idx & 63;
    float s = sab[o];
    for (int j = 0; j < 64; ++j) s += Z[bq * 64 + j] * saw[j * 64 + o];
    const float att = 1.0f / (1.0f + expf(-s));
    zc[bq * 192 + layer * 64 + o] = rb[bq * 64 + o] * att;
  }
}

// ---------------------------------------------------------------------------
// Kernel 9: augment_adj via WMMA: out = ((A+I)@(A+I)) * (1-I), batched.
// Clamped addresses + multiplicative 0/1 masks (branch-free inner loop).
// ---------------------------------------------------------------------------
__global__ void k_augadj_wmma(const float* __restrict__ adp,
                              float* __restrict__ out, int m) {
  WAVE_SETUP();
  const int tt = (m + 15) >> 4;
  const int tid = blockIdx.x * 8 + wave;
  if (tid >= tt * tt) return;
  const int tm = tid / tt, tn = tid % tt;
  const int b = blockIdx.y;
  const float* A = adp + (size_t)b * m * m;
  float* O = out + (size_t)b * m * m;

  v8f acc;
  for (int v = 0; v < 8; ++v) acc[v] = 0.0f;
  const int ar = tm * 16 + lm;
  const int bc = tn * 16 + lm;
  const float am = (ar < m) ? 1.0f : 0.0f;
  const float bmk = (bc < m) ? 1.0f : 0.0f;
  const int arc = (ar < m) ? ar : (m - 1);
  const int bcc = (bc < m) ? bc : (m - 1);
  const float* Arow = A + (size_t)arc * m;
  for (int kb = 0; kb < m; kb += 4) {
    const int k0 = kb + 2 * half;
    const int k1 = k0 + 1;
    const float km0 = (k0 < m) ? 1.0f : 0.0f;
    const float km1 = (k1 < m) ? 1.0f : 0.0f;
    const int k0c = (k0 < m) ? k0 : (m - 1);
    const int k1c = (k1 < m) ? k1 : (m - 1);
    const float a0 = Arow[k0c] + ((arc == k0c) ? 1.0f : 0.0f);
    const float a1 = Arow[k1c] + ((arc == k1c) ? 1.0f : 0.0f);
    const float b0 = A[(size_t)k0c * m + bcc] + ((k0c == bcc) ? 1.0f : 0.0f);
    const float b1 = A[(size_t)k1c * m + bcc] + ((k1c == bcc) ? 1.0f : 0.0f);
    v2f a, bf;
    a.x = a0 * (am * km0);
    a.y = a1 * (am * km1);
    bf.x = b0 * (bmk * km0);
    bf.y = b1 * (bmk * km1);
    acc = wmma_f32x4(a, bf, acc);
  }
  for (int v = 0; v < 8; ++v) {
    const int r = tm * 16 + v + 8 * half;
    const int c = tn * 16 + lm;
    if (r < m && c < m) O[(size_t)r * m + c] = (r == c) ? 0.0f : acc[v];
  }
}

// ---------------------------------------------------------------------------
// Kernel 10: head MLP with batch-stat BN, single block.
// ---------------------------------------------------------------------------
__global__ void k_head(const float* __restrict__ zc,
                       const float* __restrict__ fcw0, const float* __restrict__ fcb0,
                       const float* __restrict__ bng0, const float* __restrict__ bnb0,
                       const float* __restrict__ fcw1, const float* __restrict__ fcb1,
                       const float* __restrict__ bng1, const float* __restrict__ bnb1,
                       const float* __restrict__ fw, const float* __restrict__ fb,
                       float* __restrict__ out) {
  const int tid = threadIdx.x;  // 256 threads
  __shared__ float Z1[64 * 64];
  __shared__ float Z2[64 * 32];
  __shared__ float mu[64], iv[64];
  // stage 1: relu(zc @ fcw0 + fcb0) -> BN
  for (int idx = tid; idx < 4096; idx += 256) {
    const int bq = idx >> 6, o = idx & 63;
    float s = fcb0[o];
    for (int j = 0; j < 192; ++j) s += zc[bq * 192 + j] * fcw0[j * 64 + o];
    Z1[idx] = fmaxf(s, 0.0f);
  }
  __syncthreads();
  if (tid < 64) {
    float s = 0.0f, s2 = 0.0f;
    for (int bq = 0; bq < 64; ++bq) {
      const float z = Z1[bq * 64 + tid];
      s += z;
      s2 += z * z;
    }
    const float mn = s * (1.0f / 64.0f);
    const float var = s2 * (1.0f / 64.0f) - mn * mn;
    mu[tid] = mn;
    iv[tid] = 1.0f / sqrtf(var + 1e-5f);
  }
  __syncthreads();
  for (int idx = tid; idx < 4096; idx += 256) {
    const int o = idx & 63;
    Z1[idx] = (Z1[idx] - mu[o]) * iv[o] * bng0[o] + bnb0[o];
  }
  __syncthreads();
  // stage 2: relu(Z1 @ fcw1 + fcb1) -> BN
  for (int idx = tid; idx < 2048; idx += 256) {
    const int bq = idx >> 5, o = idx & 31;
    float s = fcb1[o];
    for (int j = 0; j < 64; ++j) s += Z1[bq * 64 + j] * fcw1[j * 32 + o];
    Z2[idx] = fmaxf(s, 0.0f);
  }
  __syncthreads();
  if (tid < 32) {
    float s = 0.0f, s2 = 0.0f;
    for (int bq = 0; bq < 64; ++bq) {
      const float z = Z2[bq * 32 + tid];
      s += z;
      s2 += z * z;
    }
    const float mn = s * (1.0f / 64.0f);
    const float var = s2 * (1.0f / 64.0f) - mn * mn;
    mu[tid] = mn;
    iv[tid] = 1.0f / sqrtf(var + 1e-5f);
  }
  __syncthreads();
  for (int idx = tid; idx < 2048; idx += 256) {
    const int o = idx & 31;
    Z2[idx] = (Z2[idx] - mu[o]) * iv[o] * bng1[o] + bnb1[o];
  }
  __syncthreads();
  // stage 3: relu(Z2 @ fw + fb) -> out (64 x 2)
  for (int idx = tid; idx < 128; idx += 256) {
    const int bq = idx >> 1, c = idx & 1;
    float s = fb[c];
    for (int j = 0; j < 32; ++j) s += Z2[bq * 32 + j] * fw[j * 2 + c];
    out[idx] = fmaxf(s, 0.0f);
  }
}

// ---------------------------------------------------------------------------
// Host-side orchestration
// ---------------------------------------------------------------------------
extern "C" void kernel_launch(void* const* d_in, const int* in_sizes, int n_in,
                              void* d_out, int out_size, void* d_ws,
                              size_t ws_size, hipStream_t stream) {
  const float* x0 = (const float*)d_in[0];
  const float* adj0 = (const float*)d_in[1];
  const float* pos0 = (const float*)d_in[2];
  // per-layer params: base 3 + 10*i : w1,w2,b2,pw,sew,seb,sbg,sbb,saw,sab
  auto P = [&](int i, int j) { return (const float*)d_in[3 + 10 * i + j]; };
  const float* fcw0 = (const float*)d_in[33];
  const float* fcb0 = (const float*)d_in[34];
  const float* bng0 = (const float*)d_in[35];
  const float* bnb0 = (const float*)d_in[36];
  const float* fcw1 = (const float*)d_in[37];
  const float* fcb1 = (const float*)d_in[38];
  const float* bng1 = (const float*)d_in[39];
  const float* bnb1 = (const float*)d_in[40];
  const float* fw = (const float*)d_in[41];
  const float* fb = (const float*)d_in[42];

  // workspace carve (floats); total ~4.36M floats (~17.4 MB)
  float* w = (float*)d_ws;
  float* Y = w;      w += 64 * 100 * 100;
  float* Hb = w;     w += 64 * 100 * 100;
  float* Apos9 = w;  w += 64 * 100 * 9;
  float* Waug = w;   w += 9 * 100 * 64;
  float* XO = w;     w += 64 * 100 * 64;
  float* RB = w;     w += 64 * 64;
  float* ZC = w;     w += 64 * 192;
  float* XP = w;     w += 3 * 64 * 50 * 64;
  float* ADJP = w;   w += 3 * 64 * 50 * 50;
  float* ADJ2 = w;   w += 3 * 64 * 50 * 50;
  float* POSP = w;   w += 3 * 64 * 50 * 100;
  int* PERM = (int*)w; w += 64 * 50;
  float* VALS = w;   w += 64 * 50;

  int n = 100, IN = 100;
  const float* cx = x0;
  const float* cadj = adj0;
  const float* cpos = pos0;

  for (int i = 0; i < 3; ++i) {
    const int m = (n + 1) / 2;
    // 1. Y = adj @ x
    {
      const int tm = (n + 15) / 16, tn = (IN + 15) / 16;
      const int tiles = tm * tn;
      dim3 g((tiles + 7) / 8, 64);
      k_adjx_wmma<<<g, 256, 0, stream>>>(cadj, cx, Y, n, IN);
    }
    // 2. h = x + Y/deg
    k_degh<<<64 * n, 128, 0, stream>>>(cadj, cx, Y, Hb, n, IN);
    // 3. Apos9 = [relu(pos @ w1) | 1]
    {
      const int tot = 64 * n * 9;
      k_posw1<<<(tot + 255) / 256, 256, 0, stream>>>(cpos, P(i, 0), Apos9, tot);
    }
    // 3b. Waug = [w2 ; b2]
    {
      const int KI64 = 8 * IN * 64;
      const int tot = 9 * IN * 64;
      k_waug<<<(tot + 255) / 256, 256, 0, stream>>>(P(i, 1), P(i, 2), Waug,
                                                    KI64, tot);
    }
    // 4. xo = fused GIN GEMM (branch-free WMMA)
    {
      const int tiles_m = (64 * n) >> 4;
      const int blocks = (tiles_m + 7) / 8;
      if (IN == 100)
        k_ginmm_wmma<100><<<blocks, 256, 0, stream>>>(Apos9, Hb, Waug, XO,
                                                      tiles_m);
      else
        k_ginmm_wmma<64><<<blocks, 256, 0, stream>>>(Apos9, Hb, Waug, XO,
                                                     tiles_m);
    }
    // 5. top-k
    k_topk<<<64, 128, 0, stream>>>(XO, P(i, 3), PERM, VALS, n, m);
    // 6. gather pooled tensors
    float* xp = XP + i * 64 * 50 * 64;
    float* adp = ADJP + i * 64 * 50 * 50;
    float* pp = POSP + i * 64 * 50 * 100;
    k_gather<<<64 * m, 128, 0, stream>>>(XO, cadj, cpos, PERM, VALS, xp, adp,
                                         pp, n, m);
    // 7. node-mean
    k_rmean<<<64, 64, 0, stream>>>(xp, RB, m);
    // 8. SERO readout -> ZC columns [64i, 64i+64)
    k_sero<<<1, 256, 0, stream>>>(RB, P(i, 4), P(i, 5), P(i, 6), P(i, 7),
                                  P(i, 8), P(i, 9), ZC, i);
    // 9. augment_adj (unused after layer 2, so skip it there)
    if (i < 2) {
      float* a2 = ADJ2 + i * 64 * 50 * 50;
      const int tt = (m + 15) / 16;
      const int tiles = tt * tt;
      dim3 g((tiles + 7) / 8, 64);
      k_augadj_wmma<<<g, 256, 0, stream>>>(adp, a2, m);
      cadj = a2;
    }
    cx = xp;
    cpos = pp;
    n = m;
    IN = 64;
  }

  // head MLP -> d_out (64 x 2 fp32)
  k_head<<<1, 256, 0, stream>>>(ZC, fcw0, fcb0, bng0, bnb0, fcw1, fcb1, bng1,
                                bnb1, fw, fb, (float*)d_out);
}